// TemporalNorm_31473520345379
// MI455X (gfx1250) — compile-verified
//
#include <hip/hip_runtime.h>

// TemporalNorm (causal rolling window W=128), B=32, T=4096, D=256, fp32.
// Memory-bound: 268 MB min HBM traffic -> ~11.5us floor at 23.3 TB/s.
// CDNA5 path: async global->LDS software pipeline (depth 24) feeding a
// 160-step LDS ring that also serves as the W=128 window history, so each
// element is read from HBM exactly once. No barriers: every lane consumes
// only LDS words it async-loaded itself; per-wave in-order ASYNCcnt waits
// are the only synchronization.

#define TN_B    32
#define TN_T    4096
#define TN_D    256
#define TN_W    128
#define TN_TILE 512
#define TN_RING 160      // > W + PF  (slot liveness) ; 160*256*4B = 160 KB LDS
#define TN_PF   24       // async pipeline depth (steps ahead)
#define TN_EPS  1e-5f

__global__ __launch_bounds__(TN_D) void temporal_norm_kernel(
    const float* __restrict__ x,
    const float* __restrict__ gamma,
    const float* __restrict__ beta,
    float* __restrict__ y)
{
    __shared__ float ring[TN_RING * TN_D];      // 160 KB -> 2 blocks / WGP

    constexpr int NT = TN_T / TN_TILE;          // 8 time tiles
    const int d    = threadIdx.x;               // channel (coalesced across lanes)
    const int b    = blockIdx.x / NT;
    const int tile = blockIdx.x % NT;
    const int t0   = tile * TN_TILE;
    const int ws   = (t0 >= TN_W) ? (t0 - TN_W) : 0;   // warm-up start
    const int tend = t0 + TN_TILE;
    const int smax = (t0 == 0) ? TN_W : t0;     // first step that subtracts

    const float wd = gamma[d];
    const float bd = beta[d];

    const float* __restrict__ xp = x + (size_t)b * TN_T * TN_D + d;
    float*       __restrict__ yp = y + (size_t)b * TN_T * TN_D + d;

    // LDS byte address of this lane's column in ring slot 0.
    const unsigned lds_col = (unsigned)(uintptr_t)ring + (unsigned)(d << 2);

    // Issue one async global->LDS B32 for (step) into (slot); per-lane
    // self-consistent: this lane later reads exactly this LDS word.
    auto issue = [&](int step, int slot) {
        unsigned lds_addr = lds_col + (unsigned)(slot << 10);   // slot*1024B
        unsigned long long ga =
            (unsigned long long)(const void*)(xp + (size_t)step * TN_D);
        asm volatile("global_load_async_to_lds_b32 %0, %1, off"
                     :: "v"(lds_addr), "v"(ga) : "memory");
    };

    // ---- prologue: fill the pipeline with PF steps --------------------
    for (int k = 0; k < TN_PF; ++k)
        issue(ws + k, k);

    float s1 = 0.0f, s2 = 0.0f;
    int slot_rd = 0;        // slot of current step s
    int slot_is = TN_PF;    // slot of step s+PF (next issue)
    int slot_sb = 0;        // slot of step s-W once subtraction starts

    for (int s = ws; s < tend; ++s) {           // all guards wave-uniform
        if (s + TN_PF < tend) {                 // steady state: keep depth PF
            issue(s + TN_PF, slot_is);
            if (++slot_is == TN_RING) slot_is = 0;
            // in-order completion: <=24 outstanding => step s is resident
            asm volatile("s_wait_asynccnt 24" ::: "memory");
        } else {                                // drain phase
            asm volatile("s_wait_asynccnt 0" ::: "memory");
        }

        float v = ring[slot_rd * TN_D + d];     // ds_load_b32
        s1 += v;
        s2  = fmaf(v, v, s2);

        if (s >= smax) {                        // sliding-window retire
            float u = ring[slot_sb * TN_D + d]; // x[s-W], still live: RING>W+PF
            s1 -= u;
            s2  = fmaf(-u, u, s2);
            if (++slot_sb == TN_RING) slot_sb = 0;
        }

        if (s >= t0) {                          // main region: emit output
            float cnt  = (s + 1 < TN_W) ? (float)(s + 1) : (float)TN_W;
            float inv  = __builtin_amdgcn_rcpf(cnt);  // exact for 128 (pow2)
            float mean = s1 * inv;
            float var  = fmaf(-mean, mean, s2 * inv);
            float r    = rsqrtf(var + TN_EPS);
            yp[(size_t)s * TN_D] = fmaf((v - mean) * r, wd, bd);
        }

        if (++slot_rd == TN_RING) slot_rd = 0;
    }
}

extern "C" void kernel_launch(void* const* d_in, const int* in_sizes, int n_in,
                              void* d_out, int out_size, void* d_ws, size_t ws_size,
                              hipStream_t stream)
{
    (void)in_sizes; (void)n_in; (void)out_size; (void)d_ws; (void)ws_size;

    const float* x     = (const float*)d_in[0];
    const float* gamma = (const float*)d_in[1];
    const float* beta  = (const float*)d_in[2];
    float*       y     = (float*)d_out;

    dim3 grid(TN_B * (TN_T / TN_TILE));   // 32 * 8 = 256 blocks
    dim3 block(TN_D);                     // 256 threads = 8 wave32
    temporal_norm_kernel<<<grid, block, 0, stream>>>(x, gamma, beta, y);
}